// ConLoss_90177133347174
// MI455X (gfx1250) — compile-verified
//
#include <hip/hip_runtime.h>
#include <hip/hip_bf16.h>

// Problem constants (from reference)
#define C_   1024
#define V_   4
#define D_   512
#define CV_  (C_ * V_)        // 4096 anchors
#define TEMP 0.07f

typedef float v2f __attribute__((ext_vector_type(2)));
typedef float v8f __attribute__((ext_vector_type(8)));
typedef int   v4i __attribute__((ext_vector_type(4)));

#define ASTRIDE 516           // 512 + 4 floats pad -> conflict-free strided b64 reads
#define BSTRIDE 68            // 64 + 4 floats pad  -> conflict-free strided b64 reads

#if __has_builtin(__builtin_amdgcn_global_load_async_to_lds_b128)
#define USE_ASYNC 1
#define WAIT_B()  asm volatile("s_wait_asynccnt 0x8" ::: "memory")
#define GCAST(p)  ((__attribute__((address_space(1))) v4i*)(void*)(p))
#define LCAST(p)  ((__attribute__((address_space(3))) v4i*)(void*)(p))
#else
#define USE_ASYNC 0
#define WAIT_B()  ((void)0)
#endif

// ---------------------------------------------------------------------------
// Main kernel: one workgroup per 16-anchor tile; 8 waves split the 8192
// candidates into 1024-row chunks. A tile staged once in LDS. B streamed
// via async global->LDS DMA (ASYNCcnt), double-buffered in 16x64 K-chunks,
// consumed by chained f32 WMMAs with a fused online logsumexp.
// ---------------------------------------------------------------------------
__global__ __launch_bounds__(256)
void conloss_logden_kernel(const float* __restrict__ src,
                           const float* __restrict__ tgt,
                           float* __restrict__ ws_logden) {
    __shared__ __align__(16) float As[16 * ASTRIDE];        // anchors tile
    __shared__ __align__(16) float Bs[8][2][16 * BSTRIDE];  // per-wave dbl buf
    __shared__ float red_m[8][16];
    __shared__ float red_s[8][16];

    const int tid  = threadIdx.x;
    const int wave = tid >> 5;
    const int lane = tid & 31;
    const int r    = lane & 15;   // A row / B column within 16x16 tile
    const int h    = lane >> 4;   // half selector -> K sub-offset
    const int m0   = blockIdx.x * 16;

    // ---- Stage A tile: tgt rows [m0, m0+16) into padded LDS ----
    {
        const float4* srcA = (const float4*)(tgt + (size_t)m0 * D_);
        #pragma unroll
        for (int it = 0; it < 8; ++it) {
            int g  = tid + 256 * it;           // float4 index, 2048 total
            int e  = g * 4;
            int rr = e / D_;
            int cc = e % D_;
            *(float4*)&As[rr * ASTRIDE + cc] = srcA[g];
        }
    }
    __syncthreads();

    const float invT = 1.0f / TEMP;
    const int abase      = r * ASTRIDE + 2 * h;  // LDS A base (k-loop pair)
    const int bbase      = r * BSTRIDE + 2 * h;  // LDS B base (k-loop pair)
    const int laneGoff   = r * D_ + 4 * h;       // staging: global float offset
    const int ldsLaneOff = r * BSTRIDE + 4 * h;  // staging: LDS float offset

    // Stage one 16-row x 64-K chunk of B (8 async b128 per wave).
    auto stage = [&](int g) {
        const int  tile = g >> 3;
        const int  cc   = g & 7;
        const int  n0   = wave * 1024 + (tile << 4);
        const float* Bb = (n0 < CV_) ? (src + (size_t)n0 * D_)
                                     : (tgt + (size_t)(n0 - CV_) * D_);
        const float* gp = Bb + laneGoff + (cc << 6);
        float*       lp = &Bs[wave][g & 1][ldsLaneOff];
#if USE_ASYNC
        #pragma unroll
        for (int i = 0; i < 8; ++i)
            __builtin_amdgcn_global_load_async_to_lds_b128(
                GCAST(gp + 8 * i), LCAST(lp + 8 * i), 0, 0);
#else
        #pragma unroll
        for (int i = 0; i < 8; ++i)
            *(float4*)(lp + 8 * i) = *(const float4*)(gp + 8 * i);
#endif
    };

    float m[8], s[8];
    #pragma unroll
    for (int j = 0; j < 8; ++j) { m[j] = -INFINITY; s[j] = 0.0f; }

    stage(0);
    v8f acc = {0.f, 0.f, 0.f, 0.f, 0.f, 0.f, 0.f, 0.f};

    // 64 N-tiles x 8 K-chunks = 512 chunks, consumed sequentially.
    for (int g = 0; g < 512; ++g) {
        if (g + 1 < 512) stage(g + 1);
        WAIT_B();                                   // chunk g resident in LDS

        const float* Bc = &Bs[wave][g & 1][0];
        const int    ck = (g & 7) << 6;             // K window base in A
        #pragma unroll
        for (int kk = 0; kk < 16; ++kk) {           // 16 WMMAs per chunk
            v2f a = *(const v2f*)&As[abase + ck + (kk << 2)];
            v2f b = *(const v2f*)&Bc[bbase + (kk << 2)];
            acc = __builtin_amdgcn_wmma_f32_16x16x4_f32(
                false, a, false, b, (short)0, acc, false, false);
        }

        if ((g & 7) == 7) {                         // full K=512 done: softmax
            #pragma unroll
            for (int j = 0; j < 8; ++j) {
                float l  = acc[j] * invT;
                float nm = fmaxf(m[j], l);
                s[j] = s[j] * __expf(m[j] - nm) + __expf(l - nm);
                m[j] = nm;
                acc[j] = 0.0f;
            }
        }
    }

    // ---- combine across the 16 lanes of each half (columns) ----
    #pragma unroll
    for (int off = 1; off <= 8; off <<= 1) {
        #pragma unroll
        for (int j = 0; j < 8; ++j) {
            float om = __shfl_xor(m[j], off, 32);
            float os = __shfl_xor(s[j], off, 32);
            float nm = fmaxf(m[j], om);
            s[j] = s[j] * __expf(m[j] - nm) + os * __expf(om - nm);
            m[j] = nm;
        }
    }
    if ((lane & 15) == 0) {                         // lanes 0 and 16
        #pragma unroll
        for (int j = 0; j < 8; ++j) {
            red_m[wave][j + 8 * h] = m[j];
            red_s[wave][j + 8 * h] = s[j];
        }
    }
    __syncthreads();

    // ---- combine across 8 waves, emit log_den per anchor ----
    if (tid < 16) {
        float M = -INFINITY, S = 0.0f;
        #pragma unroll
        for (int w = 0; w < 8; ++w) {
            float om = red_m[w][tid];
            float os = red_s[w][tid];
            float nm = fmaxf(M, om);
            S = S * __expf(M - nm) + os * __expf(om - nm);
            M = nm;
        }
        ws_logden[m0 + tid] = M + __logf(S);
    }
}

// ---------------------------------------------------------------------------
// Positive term: pos[k,v] = dot(tgt[k,v], sum_j src[k,j]) / (T*V)
// ---------------------------------------------------------------------------
__global__ __launch_bounds__(256)
void conloss_pos_kernel(const float* __restrict__ src,
                        const float* __restrict__ tgt,
                        float* __restrict__ ws_pos) {
    const int tid  = threadIdx.x;
    const int wave = tid >> 5;
    const int lane = tid & 31;
    const int gw   = blockIdx.x * 8 + wave;    // 0..127
    const float scale = 1.0f / (TEMP * (float)V_);

    for (int i = 0; i < 8; ++i) {
        const int k = gw * 8 + i;              // k-group 0..1023
        const float* sb = src + (size_t)k * 4 * D_;
        const float* tb = tgt + (size_t)k * 4 * D_;
        float p0 = 0.f, p1 = 0.f, p2 = 0.f, p3 = 0.f;
        #pragma unroll 4
        for (int t = 0; t < 16; ++t) {
            int d = lane + 32 * t;
            float ss = sb[0 * D_ + d] + sb[1 * D_ + d] + sb[2 * D_ + d] + sb[3 * D_ + d];
            p0 += tb[0 * D_ + d] * ss;
            p1 += tb[1 * D_ + d] * ss;
            p2 += tb[2 * D_ + d] * ss;
            p3 += tb[3 * D_ + d] * ss;
        }
        #pragma unroll
        for (int off = 1; off <= 16; off <<= 1) {
            p0 += __shfl_xor(p0, off, 32);
            p1 += __shfl_xor(p1, off, 32);
            p2 += __shfl_xor(p2, off, 32);
            p3 += __shfl_xor(p3, off, 32);
        }
        if (lane == 0) {
            ws_pos[k * 4 + 0] = p0 * scale;
            ws_pos[k * 4 + 1] = p1 * scale;
            ws_pos[k * 4 + 2] = p2 * scale;
            ws_pos[k * 4 + 3] = p3 * scale;
        }
    }
}

// ---------------------------------------------------------------------------
// Deterministic final reduction: loss = sum_i (log_den[i] - pos[i])
// ---------------------------------------------------------------------------
__global__ __launch_bounds__(256)
void conloss_reduce_kernel(const float* __restrict__ ws, float* __restrict__ out) {
    __shared__ float sm[256];
    const int tid = threadIdx.x;
    float acc = 0.0f;
    for (int i = tid; i < CV_; i += 256)
        acc += ws[i] - ws[CV_ + i];
    sm[tid] = acc;
    __syncthreads();
    for (int st = 128; st > 0; st >>= 1) {
        if (tid < st) sm[tid] += sm[tid + st];
        __syncthreads();
    }
    if (tid == 0) out[0] = sm[0];
}

extern "C" void kernel_launch(void* const* d_in, const int* in_sizes, int n_in,
                              void* d_out, int out_size, void* d_ws, size_t ws_size,
                              hipStream_t stream) {
    (void)in_sizes; (void)n_in; (void)out_size; (void)ws_size;
    const float* src = (const float*)d_in[0];   // [C, V, D] f32
    const float* tgt = (const float*)d_in[1];   // [C, V, D] f32
    float* ws = (float*)d_ws;                   // [0..4095]=log_den, [4096..8191]=pos

    conloss_logden_kernel<<<CV_ / 16, 256, 0, stream>>>(src, tgt, ws);
    conloss_pos_kernel<<<16, 256, 0, stream>>>(src, tgt, ws + CV_);
    conloss_reduce_kernel<<<1, 256, 0, stream>>>(ws, (float*)d_out);
}